// DSTSolver_31353261261236
// MI455X (gfx1250) — compile-verified
//

#include <hip/hip_runtime.h>
#include <hip/hip_bf16.h>
#include <math.h>

// ---------------------------------------------------------------------------
// QG elliptic solver via DST-I-as-GEMM on CDNA5 WMMA.
//   psi_modes = S * ((S * f * S) / H) * S,  S[j,k] = (1/32) sin(pi (j+1)(k+1)/2048)
// S row/col 2047 are exactly zero, so all operands are zero-padded to 2048x2048
// and the GEMMs are clean 2048^3 f32 WMMA GEMMs (16x16x4 shape).
// GEMM staging: double-buffered LDS, async global->LDS (ASYNCcnt) when the
// toolchain exposes the gfx1250 builtins, register-pipelined fallback otherwise.
// Workspace requirement: 13 * 2048*2048*4 B + 32 KB ~= 218 MB.
// ---------------------------------------------------------------------------

typedef __attribute__((ext_vector_type(2))) float v2f;
typedef __attribute__((ext_vector_type(8))) float v8f;

#define NF 2048               // padded transform size
#define NV 2047               // valid size
#define NP 2049               // padded psi size
static constexpr size_t N2  = (size_t)NF * NF;
static constexpr size_t NV2 = (size_t)NV * NV;
static constexpr size_t NP2 = (size_t)NP * NP;

#if __has_builtin(__builtin_amdgcn_global_load_async_to_lds_b128) && \
    __has_builtin(__builtin_amdgcn_s_wait_asynccnt)
#define QG_ASYNC_LDS 1
typedef __attribute__((ext_vector_type(4))) int qg_v4i;
typedef __attribute__((address_space(1))) qg_v4i* qg_g4p;   // global int4*
typedef __attribute__((address_space(3))) qg_v4i* qg_l4p;   // LDS int4*
#else
#define QG_ASYNC_LDS 0
#endif

// ---------------- build DST-I matrix ---------------------------------------
__global__ void qg_build_S(float* __restrict__ S) {
    size_t idx = (size_t)blockIdx.x * blockDim.x + threadIdx.x;
    if (idx >= N2) return;
    int j = (int)(idx >> 11);
    int k = (int)(idx & 2047);
    // sin(pi*(j+1)*(k+1)/2048): period in the integer product is 4096
    int t = (int)(((long long)(j + 1) * (long long)(k + 1)) & 4095LL);
    float ang = (float)t * (float)(M_PI / 2048.0);
    S[idx] = 0.03125f * sinf(ang);   // 1/32 = sqrt(2/2048)
}

// ---------------- layer -> mode, with zero padding to 2048 -----------------
__global__ void qg_mode_combine(const float* __restrict__ q,
                                const float* __restrict__ Cl2m,
                                float* __restrict__ F) {
    size_t idx = (size_t)blockIdx.x * blockDim.x + threadIdx.x;
    if (idx >= N2) return;
    int x = (int)(idx >> 11);
    int y = (int)(idx & 2047);
    float v[4] = {0.f, 0.f, 0.f, 0.f};
    if (x < NV && y < NV) {
        size_t qi = (size_t)x * NV + y;
        float q0 = q[0 * NV2 + qi];
        float q1 = q[1 * NV2 + qi];
        float q2 = q[2 * NV2 + qi];
        float q3 = q[3 * NV2 + qi];
#pragma unroll
        for (int m = 0; m < 4; ++m)
            v[m] = Cl2m[m * 4 + 0] * q0 + Cl2m[m * 4 + 1] * q1 +
                   Cl2m[m * 4 + 2] * q2 + Cl2m[m * 4 + 3] * q3;
    }
#pragma unroll
    for (int m = 0; m < 4; ++m) F[(size_t)m * N2 + idx] = v[m];
}

// ---------------- 2048^3 f32 WMMA GEMM (z-batched over 4 modes) ------------
// C[m] = A[ma] * B[mb]   (optionally elementwise-divided by H[m] on store)
// Block tile 128x128, 8 waves (4x2) of 32x64, K staged in chunks of 16,
// double-buffered through LDS.
__global__ void __launch_bounds__(256)
qg_gemm2048(const float* __restrict__ A, const float* __restrict__ B,
            float* __restrict__ C, const float* __restrict__ H,
            int aBatched, int bBatched) {
    const int m = blockIdx.z;
    A += (size_t)(aBatched ? m : 0) * N2;
    B += (size_t)(bBatched ? m : 0) * N2;
    C += (size_t)m * N2;
    const float* Hm = H ? (H + (size_t)m * NV2) : nullptr;

    __shared__ float As[2][128][20];   // stride 20: float4-aligned, conflict-free reads
    __shared__ float Bs[2][16][132];   // stride 132: float4-aligned stores, linear reads

    const int tid     = threadIdx.x;
    const int lane    = tid & 31;
    const int w       = tid >> 5;
    const int waveRow = w >> 1;       // 0..3  -> 32-row slice
    const int waveCol = w & 1;        // 0..1  -> 64-col slice
    const int lm      = lane & 15;    // M (A) / N (B) index within 16
    const int kh2     = (lane >> 4) << 1;  // K half: 0 or 2
    const int mOff    = (lane >> 4) * 8;   // C/D: M offset for upper lanes

    const int rowBase = blockIdx.y * 128;
    const int colBase = blockIdx.x * 128;

    // per-thread staging coordinates (A: 512 float4, B: 512 float4, 2 each)
    const int ar0 = (tid * 2 + 0) >> 2,  ac0 = ((tid * 2 + 0) & 3) * 4;
    const int ar1 = (tid * 2 + 1) >> 2,  ac1 = ((tid * 2 + 1) & 3) * 4;
    const int bk0 = (tid * 2 + 0) >> 5,  bc0 = ((tid * 2 + 0) & 31) * 4;
    const int bk1 = (tid * 2 + 1) >> 5,  bc1 = ((tid * 2 + 1) & 31) * 4;

    v8f acc[2][4];
#pragma unroll
    for (int i = 0; i < 2; ++i)
#pragma unroll
        for (int j = 0; j < 4; ++j) acc[i][j] = (v8f)0.0f;

#if QG_ASYNC_LDS
    // ---- async global->LDS double-buffered pipeline ----
    {
        // preload stage 0 into buffer 0
        __builtin_amdgcn_global_load_async_to_lds_b128(
            (qg_g4p)(A + (size_t)(rowBase + ar0) * NF + 0 + ac0),
            (qg_l4p)&As[0][ar0][ac0], 0, 0);
        __builtin_amdgcn_global_load_async_to_lds_b128(
            (qg_g4p)(A + (size_t)(rowBase + ar1) * NF + 0 + ac1),
            (qg_l4p)&As[0][ar1][ac1], 0, 0);
        __builtin_amdgcn_global_load_async_to_lds_b128(
            (qg_g4p)(B + (size_t)(0 + bk0) * NF + colBase + bc0),
            (qg_l4p)&Bs[0][bk0][bc0], 0, 0);
        __builtin_amdgcn_global_load_async_to_lds_b128(
            (qg_g4p)(B + (size_t)(0 + bk1) * NF + colBase + bc1),
            (qg_l4p)&Bs[0][bk1][bc1], 0, 0);
        __builtin_amdgcn_s_wait_asynccnt(0);
        __syncthreads();
    }
#else
    // ---- register-pipelined fallback: preload stage 0 ----
    float4 va0, va1, vb0, vb1;
    va0 = *(const float4*)(A + (size_t)(rowBase + ar0) * NF + 0 + ac0);
    va1 = *(const float4*)(A + (size_t)(rowBase + ar1) * NF + 0 + ac1);
    vb0 = *(const float4*)(B + (size_t)(0 + bk0) * NF + colBase + bc0);
    vb1 = *(const float4*)(B + (size_t)(0 + bk1) * NF + colBase + bc1);
    *(float4*)&As[0][ar0][ac0] = va0;
    *(float4*)&As[0][ar1][ac1] = va1;
    *(float4*)&Bs[0][bk0][bc0] = vb0;
    *(float4*)&Bs[0][bk1][bc1] = vb1;
    __syncthreads();
#endif

    for (int kb = 0; kb < NF; kb += 16) {
        const int cur = (kb >> 4) & 1;
        const int nxt = cur ^ 1;
        const int kn  = kb + 16;
        const bool more = (kn < NF);

#if QG_ASYNC_LDS
        if (more) {
            __builtin_amdgcn_global_load_async_to_lds_b128(
                (qg_g4p)(A + (size_t)(rowBase + ar0) * NF + kn + ac0),
                (qg_l4p)&As[nxt][ar0][ac0], 0, 0);
            __builtin_amdgcn_global_load_async_to_lds_b128(
                (qg_g4p)(A + (size_t)(rowBase + ar1) * NF + kn + ac1),
                (qg_l4p)&As[nxt][ar1][ac1], 0, 0);
            __builtin_amdgcn_global_load_async_to_lds_b128(
                (qg_g4p)(B + (size_t)(kn + bk0) * NF + colBase + bc0),
                (qg_l4p)&Bs[nxt][bk0][bc0], 0, 0);
            __builtin_amdgcn_global_load_async_to_lds_b128(
                (qg_g4p)(B + (size_t)(kn + bk1) * NF + colBase + bc1),
                (qg_l4p)&Bs[nxt][bk1][bc1], 0, 0);
        }
#else
        if (more) {   // issue global loads early; latency overlaps WMMA loop
            va0 = *(const float4*)(A + (size_t)(rowBase + ar0) * NF + kn + ac0);
            va1 = *(const float4*)(A + (size_t)(rowBase + ar1) * NF + kn + ac1);
            vb0 = *(const float4*)(B + (size_t)(kn + bk0) * NF + colBase + bc0);
            vb1 = *(const float4*)(B + (size_t)(kn + bk1) * NF + colBase + bc1);
        }
#endif

#pragma unroll
        for (int kk = 0; kk < 16; kk += 4) {
            const int kh = kk + kh2;
            v2f a0, a1;
            a0.x = As[cur][waveRow * 32 + lm][kh];
            a0.y = As[cur][waveRow * 32 + lm][kh + 1];
            a1.x = As[cur][waveRow * 32 + 16 + lm][kh];
            a1.y = As[cur][waveRow * 32 + 16 + lm][kh + 1];
            v2f bf[4];
#pragma unroll
            for (int j = 0; j < 4; ++j) {
                const int n = waveCol * 64 + j * 16 + lm;
                bf[j].x = Bs[cur][kh][n];
                bf[j].y = Bs[cur][kh + 1][n];
            }
#pragma unroll
            for (int j = 0; j < 4; ++j) {
                acc[0][j] = __builtin_amdgcn_wmma_f32_16x16x4_f32(
                    false, a0, false, bf[j], (short)0, acc[0][j], false, false);
                acc[1][j] = __builtin_amdgcn_wmma_f32_16x16x4_f32(
                    false, a1, false, bf[j], (short)0, acc[1][j], false, false);
            }
        }

#if QG_ASYNC_LDS
        __builtin_amdgcn_s_wait_asynccnt(0);
        __syncthreads();
#else
        if (more) {   // commit next tile (nobody reads buffer `nxt` right now)
            *(float4*)&As[nxt][ar0][ac0] = va0;
            *(float4*)&As[nxt][ar1][ac1] = va1;
            *(float4*)&Bs[nxt][bk0][bc0] = vb0;
            *(float4*)&Bs[nxt][bk1][bc1] = vb1;
        }
        __syncthreads();
#endif
    }

    // store (C/D layout: VGPR r holds M = mOff + r, N = lm)
#pragma unroll
    for (int i = 0; i < 2; ++i) {
#pragma unroll
        for (int j = 0; j < 4; ++j) {
            const int col = colBase + waveCol * 64 + j * 16 + lm;
#pragma unroll
            for (int r = 0; r < 8; ++r) {
                const int row = rowBase + waveRow * 32 + i * 16 + mOff + r;
                float v = acc[i][j][r];
                if (Hm) {
                    v = (row < NV && col < NV) ? v / Hm[(size_t)row * NV + col] : 0.0f;
                }
                C[(size_t)row * NF + col] = v;
            }
        }
    }
}

// ---------------- deterministic interior-sum reduction ---------------------
__global__ void qg_reduce_rows(const float* __restrict__ P, float* __restrict__ partial) {
    const int row = blockIdx.x;   // 0..2046
    const int m   = blockIdx.y;
    const float* p = P + (size_t)m * N2 + (size_t)row * NF;
    float s = 0.f;
    for (int c = threadIdx.x; c < NV; c += blockDim.x) s += p[c];
    __shared__ float sm[256];
    sm[threadIdx.x] = s;
    __syncthreads();
    for (int off = 128; off > 0; off >>= 1) {
        if (threadIdx.x < off) sm[threadIdx.x] += sm[threadIdx.x + off];
        __syncthreads();
    }
    if (threadIdx.x == 0) partial[m * 2048 + row] = sm[0];
}

__global__ void qg_reduce_final(const float* __restrict__ partial, float* __restrict__ sums) {
    const int m = blockIdx.x;
    float s = 0.f;
    for (int r = threadIdx.x; r < NV; r += blockDim.x) s += partial[m * 2048 + r];
    __shared__ float sm[256];
    sm[threadIdx.x] = s;
    __syncthreads();
    for (int off = 128; off > 0; off >>= 1) {
        if (threadIdx.x < off) sm[threadIdx.x] += sm[threadIdx.x + off];
        __syncthreads();
    }
    if (threadIdx.x == 0) sums[m] = sm[0];
}

// ---------------- homsol correction + mode -> layer ------------------------
__global__ void qg_finalize(const float* __restrict__ PSI,     // [4,2048,2048] interior
                            const float* __restrict__ homsol,  // [4,2049,2049]
                            const float* __restrict__ sums,    // [4]
                            const float* __restrict__ homsol_mean,  // [4]
                            const float* __restrict__ Cm2l,    // [4,4]
                            float* __restrict__ out) {         // [4,2049,2049]
    size_t idx = (size_t)blockIdx.x * blockDim.x + threadIdx.x;
    if (idx >= NP2) return;
    const int x = (int)(idx / NP);
    const int y = (int)(idx % NP);
    const bool interior = (x >= 1) && (x <= NV) && (y >= 1) && (y <= NV);
    float vals[4];
#pragma unroll
    for (int mm = 0; mm < 4; ++mm) {
        // psi_modes_mean = sum(interior) / 2048^2 ; alpha = -mean / homsol_mean
        const float mean  = sums[mm] * (1.0f / (2048.0f * 2048.0f));
        const float alpha = -mean / homsol_mean[mm];
        float v = alpha * homsol[(size_t)mm * NP2 + idx];
        if (interior)
            v += PSI[(size_t)mm * N2 + (size_t)(x - 1) * NF + (y - 1)];
        vals[mm] = v;
    }
#pragma unroll
    for (int l = 0; l < 4; ++l) {
        float s = Cm2l[l * 4 + 0] * vals[0] + Cm2l[l * 4 + 1] * vals[1] +
                  Cm2l[l * 4 + 2] * vals[2] + Cm2l[l * 4 + 3] * vals[3];
        out[(size_t)l * NP2 + idx] = s;
    }
}

// ---------------------------------------------------------------------------
extern "C" void kernel_launch(void* const* d_in, const int* in_sizes, int n_in,
                              void* d_out, int out_size, void* d_ws, size_t ws_size,
                              hipStream_t stream) {
    (void)in_sizes; (void)n_in; (void)out_size; (void)ws_size;
    const float* q           = (const float*)d_in[0];  // [4,2047,2047]
    const float* Cl2m        = (const float*)d_in[1];  // [4,4]
    const float* Cm2l        = (const float*)d_in[2];  // [4,4]
    const float* Hmat        = (const float*)d_in[3];  // [4,2047,2047]
    const float* homsol      = (const float*)d_in[4];  // [4,2049,2049]
    const float* homsol_mean = (const float*)d_in[5];  // [4]
    float* out = (float*)d_out;

    char* ws = (char*)d_ws;
    const size_t N2b = N2 * sizeof(float);              // 16.78 MB
    float* S  = (float*)(ws);                           // [2048,2048]
    float* F  = (float*)(ws + N2b * 1);                 // [4,2048,2048]
    float* T1 = (float*)(ws + N2b * 5);                 // [4,2048,2048]
    float* G  = (float*)(ws + N2b * 9);                 // [4,2048,2048]
    float* partial = (float*)(ws + N2b * 13);           // [4,2048]
    float* sums    = partial + 4 * 2048;                // [4]

    const int nElem = (int)(N2 / 256);
    qg_build_S<<<nElem, 256, 0, stream>>>(S);
    qg_mode_combine<<<nElem, 256, 0, stream>>>(q, Cl2m, F);

    dim3 gg(16, 16, 4), gb(256);
    // T1 = S * F          (DST along x)
    qg_gemm2048<<<gg, gb, 0, stream>>>(S, F, T1, nullptr, 0, 1);
    // G  = (T1 * S) / H   (DST along y, spectral divide fused)
    qg_gemm2048<<<gg, gb, 0, stream>>>(T1, S, G, Hmat, 1, 0);
    // F  = S * G          (inverse DST along x)
    qg_gemm2048<<<gg, gb, 0, stream>>>(S, G, F, nullptr, 0, 1);
    // T1 = F * S          (inverse DST along y -> psi interior)
    qg_gemm2048<<<gg, gb, 0, stream>>>(F, S, T1, nullptr, 1, 0);

    qg_reduce_rows<<<dim3(NV, 4), 256, 0, stream>>>(T1, partial);
    qg_reduce_final<<<4, 256, 0, stream>>>(partial, sums);

    const int npBlocks = (int)((NP2 + 255) / 256);
    qg_finalize<<<npBlocks, 256, 0, stream>>>(T1, homsol, sums, homsol_mean, Cm2l, out);
}